// Yolo_50611894616705
// MI455X (gfx1250) — compile-verified
//
#include <hip/hip_runtime.h>
#include <math.h>

// ---------------------------------------------------------------------------
// YOLO decode for MI455X (gfx1250).
// Pure bandwidth kernel: ~248MB in + ~266MB out => ~22us floor at 23.3 TB/s.
//  - per-channel coalesced NT load streams (data is read exactly once)
//  - block-uniform index decomposition in SALU (one wrap max per block),
//    lanes only do cndmask selects -> minimal VALU alongside memory
//  - LDS staging makes each block's 15-float outputs one contiguous 15360B
//    slab, shipped with GLOBAL_STORE_ASYNC_FROM_LDS_B128 (ASYNCcnt path)
// ---------------------------------------------------------------------------

namespace {

constexpr int   G_      = 152;
constexpr int   GG_     = G_ * G_;      // 23104
constexpr int   NUMA_   = 3;
constexpr int   CPA_    = 14;           // 7 + NC channels per anchor
constexpr int   QOUT_   = 15;
constexpr int   TILE_   = 256;
constexpr float STRIDE_ = 4.0f;         // 608 / 152

// Builtin signature (confirmed by clang diagnostic): (v4i AS1*, v4i AS3*, Ii, Ii)
typedef int v4i_ __attribute__((vector_size(16)));
typedef __attribute__((address_space(1))) v4i_* g_v4i_ptr;
typedef __attribute__((address_space(3))) v4i_* l_v4i_ptr;

__device__ __forceinline__ float fsigmoid(float v) {
    return __builtin_amdgcn_rcpf(1.0f + __expf(-v));
}

// Scrambled cls channel: flat f = 3*(q-8)+a -> src anchor f/7, class f%7.
__device__ __forceinline__ constexpr int chan_of(int q, int a) {
    const int f    = 3 * (q - 8) + a;
    const int asrc = (f >= 7) + (f >= 14);          // f < 21
    return asrc * CPA_ + 7 + (f - 7 * asrc);
}

__global__ __launch_bounds__(TILE_) void yolo_decode_kernel(
    const float* __restrict__ x,        // (64, 42, 152, 152)
    const float* __restrict__ anchors,  // (3, 2)
    float* __restrict__ out)            // (64, 3*152*152, 15)
{
    __shared__ float tile_lds[TILE_ * QOUT_];   // 15360 B == block's output slab

    const int tid = threadIdx.x;
    const int B0  = blockIdx.x * TILE_;         // uniform

    // ---- block-uniform decomposition (SALU) --------------------------------
    // Within a block, s = s0 + tid wraps the GG_ plane at most once.
    const int ba0 = B0 / GG_;
    const int s0  = B0 - ba0 * GG_;
    const int b0  = ba0 / NUMA_;
    const int a0  = ba0 - b0 * NUMA_;
    const int a1  = (a0 == NUMA_ - 1) ? 0 : a0 + 1;        // anchor after wrap
    const int b1  = (a0 == NUMA_ - 1) ? b0 + 1 : b0;       // batch  after wrap

    const int cb0 = b0 * (NUMA_ * CPA_) * GG_;             // batch channel base
    const int cb1 = b1 * (NUMA_ * CPA_) * GG_;
    const int pa0 = cb0 + a0 * (CPA_ * GG_);               // anchor pred base
    const int pa1 = cb1 + a1 * (CPA_ * GG_);

    // Uniform anchor w/h for both candidate anchors (scalar loads).
    const float aw0 = anchors[2 * a0 + 0], ah0 = anchors[2 * a0 + 1];
    const float aw1 = anchors[2 * a1 + 0], ah1 = anchors[2 * a1 + 1];

    // ---- per-lane part: one compare + selects ------------------------------
    const int  st   = s0 + tid;
    const bool wrap = st >= GG_;
    const int  s    = wrap ? st - GG_ : st;
    const int  gy   = s / G_;                  // the one per-lane division left
    const int  gx   = s - gy * G_;

    const int aoff = (wrap ? pa1 : pa0) + s;   // element offset of pred channel 0
    const float anc_w = wrap ? aw1 : aw0;
    const float anc_h = wrap ? ah1 : ah0;

    // 7 coalesced NT load streams for this anchor's pred channels.
    const float y0 = __builtin_nontemporal_load(x + aoff + 0 * GG_);
    const float y1 = __builtin_nontemporal_load(x + aoff + 1 * GG_);
    const float y2 = __builtin_nontemporal_load(x + aoff + 2 * GG_);
    const float y3 = __builtin_nontemporal_load(x + aoff + 3 * GG_);
    const float im = __builtin_nontemporal_load(x + aoff + 4 * GG_);
    const float re = __builtin_nontemporal_load(x + aoff + 5 * GG_);
    const float y6 = __builtin_nontemporal_load(x + aoff + 6 * GG_);

    float* q15 = &tile_lds[tid * QOUT_];   // stride 15 coprime to 64 banks
    q15[0] = im;
    q15[1] = re;
    q15[2] = atanf(im * __builtin_amdgcn_rcpf(re));   // atan(im/re)
    q15[3] = fsigmoid(y6);
    q15[4] = floorf((fsigmoid(y0) + (float)gx) * STRIDE_);
    q15[5] = floorf((fsigmoid(y1) + (float)gy) * STRIDE_);
    q15[6] = __expf(y2) * anc_w;
    q15[7] = __expf(y3) * anc_h;

    // cls: scalar per-q offset tables for the two candidate anchors, lane-select.
#pragma unroll
    for (int q = 8; q < QOUT_; ++q) {
        const int off0 = cb0 + chan_of(q, a0) * GG_;   // uniform (SALU)
        const int off1 = cb1 + chan_of(q, a1) * GG_;   // uniform (SALU)
        q15[q] = __builtin_nontemporal_load(x + (wrap ? off1 : off0) + s);
    }

    __syncthreads();   // all DS writes visible before the LDS->global transfer

    // ---- ship the contiguous 15360B tile to global -------------------------
#if __has_builtin(__builtin_amdgcn_global_store_async_from_lds_b128)
    {
        char* gbase = (char*)(out + (size_t)blockIdx.x * (TILE_ * QOUT_));
        constexpr int NXFER = TILE_ * QOUT_ * 4 / 16;   // 960 x 16B transfers
#pragma unroll
        for (int j = 0; j < (NXFER + TILE_ - 1) / TILE_; ++j) {
            const int t = j * TILE_ + tid;
            if (t < NXFER) {
                __builtin_amdgcn_global_store_async_from_lds_b128(
                    (g_v4i_ptr)(gbase + (size_t)t * 16),
                    (l_v4i_ptr)((char*)tile_lds + (size_t)t * 16),
                    /*offset=*/0, /*cpol=TH_STORE_NT*/1);
            }
        }
#if __has_builtin(__builtin_amdgcn_s_wait_asynccnt)
        __builtin_amdgcn_s_wait_asynccnt(0);
#endif
    }
#else
    {
        float* gbase = out + (size_t)blockIdx.x * (TILE_ * QOUT_);
#pragma unroll
        for (int q = 0; q < QOUT_; ++q)
            gbase[q * TILE_ + tid] = tile_lds[q * TILE_ + tid];
    }
#endif
}

} // anonymous namespace

extern "C" void kernel_launch(void* const* d_in, const int* in_sizes, int n_in,
                              void* d_out, int out_size, void* d_ws, size_t ws_size,
                              hipStream_t stream) {
    const float* x       = (const float*)d_in[0];
    const float* anchors = (const float*)d_in[1];
    float*       out     = (float*)d_out;

    const int npos   = out_size / QOUT_;          // 4,435,968
    const int blocks = npos / TILE_;              // 17,328 (exact division)

    yolo_decode_kernel<<<blocks, TILE_, 0, stream>>>(x, anchors, out);
}